// DWTEnhancement_14577119003551
// MI455X (gfx1250) — compile-verified
//
#include <hip/hip_runtime.h>
#include <hip/hip_bf16.h>

// ---------------------------------------------------------------------------
// DWT enhancement (db4, 9 levels, soft threshold, inverse) for MI455X gfx1250.
// Memory-bound (~0.8 GB traffic => ~35us at 23.3 TB/s), so keep full FP32 and
// use V_WMMA_F32_16X16X4_F32 for the analysis filter bank:
//   16 outputs x 16 signals = W(16x40 banded) x X(40x16), K accumulated as
//   10 x wmma 16x16x4 chains (x2 filters). N dim = 16 contiguous features.
// ---------------------------------------------------------------------------

typedef __attribute__((ext_vector_type(2))) float v2f;
typedef __attribute__((ext_vector_type(8))) float v8f;

__device__ const float DEC_LO[8] = {
    -0.010597401784997278f,  0.032883011666982945f,  0.030841381835986965f,
    -0.18703481171888114f,  -0.02798376941698385f,   0.6308807679295904f,
     0.7148465705525415f,    0.23037781330885523f};
__device__ const float DEC_HI[8] = {
    -0.23037781330885523f,   0.7148465705525415f,   -0.6308807679295904f,
    -0.02798376941698385f,   0.18703481171888114f,   0.030841381835986965f,
    -0.032883011666982945f, -0.010597401784997278f};

__device__ __forceinline__ float tap8(const float* h, int t) {
  return (t >= 0 && t < 8) ? h[t] : 0.0f;
}

__device__ __forceinline__ v8f wmma_f32(v2f a, v2f b, v8f c) {
  // D = A(16x4 f32) * B(4x16 f32) + C(16x16 f32)
  return __builtin_amdgcn_wmma_f32_16x16x4_f32(
      false, a, false, b, (short)0, c, false, false);
}

// ------------------------- forward DWT level (WMMA) -------------------------
template <int LEVEL0>
__global__ __launch_bounds__(256) void dwt_fwd_wmma(
    const float* __restrict__ src, float* __restrict__ a_out,
    float* __restrict__ d_out, int n_in, int n_out,
    int pitch_in, int pitch_out) {
  const int tile = blockIdx.x;          // 0..511
  const int lane = threadIdx.x & 31;
  const int wave = threadIdx.x >> 5;    // 0..7
  const int nl   = lane & 15;           // A-row M / B-col N / D-col N
  const int kh   = lane >> 4;

  // Banded operator fragments: W[m][k] = h[7 + 2m - k]
  v2f Aa[10], Ad[10];
#pragma unroll
  for (int kb = 0; kb < 10; ++kb) {
    int k0 = 4 * kb + 2 * kh;
    int t0 = 7 + 2 * nl - k0;
    Aa[kb].x = tap8(DEC_LO, t0);
    Aa[kb].y = tap8(DEC_LO, t0 - 1);
    Ad[kb].x = tap8(DEC_HI, t0);
    Ad[kb].y = tap8(DEC_HI, t0 - 1);
  }

  const int b  = tile >> 2;
  const int f0 = (tile & 3) << 4;
  const size_t base0 = ((size_t)b * 4096) * 64 + (size_t)(f0 + nl);
  const size_t basel = (size_t)tile * (size_t)pitch_in + (size_t)nl;
  const int pmax  = 2 * n_in + 6;
  const int mirr  = 2 * n_in - 1;

  // Branchless symmetric-pad reflect: xe[p] = x[s]
  auto xptr = [&](int p) -> const float* {
    p = (p > pmax) ? pmax : p;
    int s = p - 7;
    s ^= (s >> 31);            // mirror at left edge: -1-s == ~s
    int r = mirr - s;          // mirror at right edge
    s = (r < s) ? r : s;
    return LEVEL0 ? (src + base0 + (size_t)s * 64)
                  : (src + basel + (size_t)s * 16);
  };

  const int nblk = (n_out + 15) >> 4;
  const size_t obase = (size_t)tile * (size_t)pitch_out + (size_t)nl;

  for (int jb = wave; jb < nblk; jb += 8) {
    const int j0 = jb << 4;
    if (jb + 8 < nblk)
      __builtin_prefetch(xptr(2 * ((jb + 8) << 4) + 1 + 2 * kh), 0, 0);

    // Issue all 20 B-fragment loads first so they overlap.
    v2f Bv[10];
#pragma unroll
    for (int kb = 0; kb < 10; ++kb) {
      int p0 = 2 * j0 + 1 + 4 * kb + 2 * kh;
      Bv[kb].x = *xptr(p0);
      Bv[kb].y = *xptr(p0 + 1);
    }

    // 20-deep WMMA chain (EXEC is all-ones here; stores diverge only after).
    v8f ca = {};
    v8f cd = {};
#pragma unroll
    for (int kb = 0; kb < 10; ++kb) {
      ca = wmma_f32(Aa[kb], Bv[kb], ca);
      cd = wmma_f32(Ad[kb], Bv[kb], cd);
    }

    // Store: full blocks take the unguarded straight-line path; only the
    // (wave-uniform) tail block pays the per-lane EXEC-mask guards.
    if (j0 + 16 <= n_out) {
#pragma unroll
      for (int r = 0; r < 8; ++r) {
        size_t o = obase + (size_t)(j0 + r + 8 * kh) * 16;
        a_out[o] = ca[r];
        d_out[o] = cd[r];
      }
    } else {
#pragma unroll
      for (int r = 0; r < 8; ++r) {
        int j = j0 + r + 8 * kh;
        if (j < n_out) {
          a_out[obase + (size_t)j * 16] = ca[r];
          d_out[obase + (size_t)j * 16] = cd[r];
        }
      }
    }
  }
}

// ----------------------- per-row std of d1 -> threshold ---------------------
__global__ __launch_bounds__(256) void d1_std(
    const float* __restrict__ d1, float* __restrict__ thr, int n_d, int pitch) {
  const int lane = threadIdx.x & 31;
  const int wave = threadIdx.x >> 5;
  const int row  = blockIdx.x * 8 + wave;  // 0..8191
  const int tile = row >> 4, l = row & 15;
  const float* p = d1 + (size_t)tile * (size_t)pitch + (size_t)l;
  float s = 0.f, s2 = 0.f;
  for (int i = lane; i < n_d; i += 32) {
    float v = p[(size_t)i * 16];
    s += v;
    s2 += v * v;
  }
#pragma unroll
  for (int m = 16; m >= 1; m >>= 1) {
    s  += __shfl_xor(s, m, 32);
    s2 += __shfl_xor(s2, m, 32);
  }
  if (lane == 0) {
    float inv  = 1.0f / (float)n_d;
    float mean = s * inv;
    float var  = fmaxf(s2 * inv - mean * mean, 0.0f);
    thr[row]   = sqrtf(var) * 4.0786676f;  // sqrt(2*ln(4096))
  }
}

// ------------------------------ soft threshold ------------------------------
__global__ __launch_bounds__(256) void soft_thr(
    float* __restrict__ buf, const float* __restrict__ thr, int n, int pitch) {
  long long idx   = (long long)blockIdx.x * blockDim.x + threadIdx.x;
  long long total = 512LL * (long long)n * 16;
  if (idx >= total) return;
  int per  = n * 16;
  int tile = (int)(idx / per);
  int rem  = (int)(idx - (long long)tile * per);
  int s = rem >> 4, l = rem & 15;
  size_t a = (size_t)tile * (size_t)pitch + (size_t)s * 16 + (size_t)l;
  float v  = buf[a];
  float t  = thr[tile * 16 + l];
  float m  = fmaxf(fabsf(v) - t, 0.0f);
  buf[a]   = copysignf(m, v);
}

// ------------------------------- inverse DWT --------------------------------
// y[i] = sum_{t == i (mod 2), 4 taps} reclo[t]*a[(i+6-t)/2] + rechi[t]*d[...]
__global__ __launch_bounds__(256) void idwt_level(
    const float* __restrict__ a_in, const float* __restrict__ d_in,
    float* __restrict__ y_out, int n_d, int pitch_a, int pitch_d, int pitch_y,
    int final_out) {
  constexpr float RL[8] = {
       0.23037781330885523f,   0.7148465705525415f,    0.6308807679295904f,
      -0.02798376941698385f,  -0.18703481171888114f,   0.030841381835986965f,
       0.032883011666982945f, -0.010597401784997278f};
  constexpr float RH[8] = {
      -0.010597401784997278f, -0.032883011666982945f,  0.030841381835986965f,
       0.18703481171888114f,  -0.02798376941698385f,  -0.6308807679295904f,
       0.7148465705525415f,   -0.23037781330885523f};

  const int y_len = 2 * n_d - 6;
  long long idx   = (long long)blockIdx.x * blockDim.x + threadIdx.x;
  long long total = 512LL * (long long)y_len * 16;
  if (idx >= total) return;
  int per  = y_len * 16;
  int tile = (int)(idx / per);
  int rem  = (int)(idx - (long long)tile * per);
  int i = rem >> 4, l = rem & 15;
  const float* ap = a_in + (size_t)tile * (size_t)pitch_a + (size_t)l;
  const float* dp = d_in + (size_t)tile * (size_t)pitch_d + (size_t)l;

  const bool odd = (i & 1) != 0;
  // Active taps selected with compile-time indices -> v_cndmask, no gathers.
  float rl[4], rh[4];
#pragma unroll
  for (int tt = 0; tt < 4; ++tt) {
    rl[tt] = odd ? RL[2 * tt + 1] : RL[2 * tt];
    rh[tt] = odd ? RH[2 * tt + 1] : RH[2 * tt];
  }

  float acc = 0.f;
  const int t0 = odd ? 1 : 0;
#pragma unroll
  for (int tt = 0; tt < 4; ++tt) {
    int t = t0 + 2 * tt;
    int k = (i + 6 - t) >> 1;
    bool ok = (unsigned)k < (unsigned)n_d;
    int kc  = ok ? k : 0;                    // clamped, always-valid address
    float w = ok ? 1.0f : 0.0f;              // zero-mask OOB contribution
    acc += w * (rl[tt] * ap[(size_t)kc * 16] + rh[tt] * dp[(size_t)kc * 16]);
  }

  if (final_out) {
    int b = tile >> 2, f0 = (tile & 3) << 4;
    y_out[(((size_t)b * 4096) + (size_t)i) * 64 + (size_t)(f0 + l)] = acc;
  } else {
    y_out[(size_t)tile * (size_t)pitch_y + (size_t)i * 16 + (size_t)l] = acc;
  }
}

// --------------------------------- driver -----------------------------------
extern "C" void kernel_launch(void* const* d_in, const int* in_sizes, int n_in,
                              void* d_out, int out_size, void* d_ws,
                              size_t ws_size, hipStream_t stream) {
  (void)in_sizes; (void)n_in; (void)out_size; (void)ws_size;
  static const int N[10]  = {4096, 2051, 1029, 518, 262, 134, 70, 38, 22, 14};
  static const int PN[10] = {4096, 2052, 1032, 520, 264, 136, 72, 40, 24, 16};

  const float* x = (const float*)d_in[0];
  float* out = (float*)d_out;
  float* ws  = (float*)d_ws;

  size_t a_off[10], d_off[10], off = 0;
  for (int l = 1; l <= 9; ++l) { a_off[l] = off; off += (size_t)512 * PN[l] * 16; }
  for (int l = 1; l <= 9; ++l) { d_off[l] = off; off += (size_t)512 * PN[l] * 16; }
  size_t thr_off = off;

  // Forward: 9 WMMA filter-bank levels.
  dwt_fwd_wmma<1><<<dim3(512), dim3(256), 0, stream>>>(
      x, ws + a_off[1], ws + d_off[1], N[0], N[1], 0, PN[1] * 16);
  for (int l = 2; l <= 9; ++l) {
    dwt_fwd_wmma<0><<<dim3(512), dim3(256), 0, stream>>>(
        ws + a_off[l - 1], ws + a_off[l], ws + d_off[l], N[l - 1], N[l],
        PN[l - 1] * 16, PN[l] * 16);
  }

  // Per-row threshold from std of first-level detail.
  d1_std<<<dim3(1024), dim3(256), 0, stream>>>(ws + d_off[1], ws + thr_off,
                                               N[1], PN[1] * 16);

  // Soft-threshold a9 and all detail bands.
  {
    long long tot = 512LL * N[9] * 16;
    soft_thr<<<dim3((unsigned)((tot + 255) / 256)), dim3(256), 0, stream>>>(
        ws + a_off[9], ws + thr_off, N[9], PN[9] * 16);
  }
  for (int l = 1; l <= 9; ++l) {
    long long tot = 512LL * N[l] * 16;
    soft_thr<<<dim3((unsigned)((tot + 255) / 256)), dim3(256), 0, stream>>>(
        ws + d_off[l], ws + thr_off, N[l], PN[l] * 16);
  }

  // Inverse: reconstruct downward, reusing a-slots as ping buffers.
  for (int l = 9; l >= 1; --l) {
    int y_len = 2 * N[l] - 6;
    long long tot = 512LL * y_len * 16;
    unsigned blocks = (unsigned)((tot + 255) / 256);
    if (l > 1) {
      idwt_level<<<dim3(blocks), dim3(256), 0, stream>>>(
          ws + a_off[l], ws + d_off[l], ws + a_off[l - 1], N[l], PN[l] * 16,
          PN[l] * 16, PN[l - 1] * 16, 0);
    } else {
      idwt_level<<<dim3(blocks), dim3(256), 0, stream>>>(
          ws + a_off[1], ws + d_off[1], out, N[1], PN[1] * 16, PN[1] * 16, 0, 1);
    }
  }
}